// GyroBNPV_69389491634908
// MI455X (gfx1250) — compile-verified
//
#include <hip/hip_runtime.h>
#include <cstdint>

#define NBLK 256
#define WPB  8           // waves per block
#define BLOCK 256

#define EPS_C      1e-12f
#define SINH_CLIP_C 30.0f
#define VAR_FLOOR_C 0.001f
#define BN_EPS_C   1e-6f
#define MAX_STEP_C 0.5f
#define TOL_C      1e-6f

typedef float v2f __attribute__((ext_vector_type(2)));
typedef float v8f __attribute__((ext_vector_type(8)));

// ---------- wave32 butterfly sum ----------
__device__ __forceinline__ float wred(float v) {
    v += __shfl_xor(v, 1, 32);
    v += __shfl_xor(v, 2, 32);
    v += __shfl_xor(v, 4, 32);
    v += __shfl_xor(v, 8, 32);
    v += __shfl_xor(v, 16, 32);
    return v;
}

// ---------- async global -> LDS copy of one 512B row (b128 per lane) ----------
__device__ __forceinline__ void async_row(uint32_t lds_byte, const float* g) {
    // VDST = per-lane LDS byte offset, VADDR = per-lane 64-bit global address
    asm volatile("global_load_async_to_lds_b128 %0, %1, off"
                 :: "v"(lds_byte), "v"(g) : "memory");
}
__device__ __forceinline__ void wait_async_le(int n) {
    if (n == 0) asm volatile("s_wait_asynccnt 0" ::: "memory");
    else        asm volatile("s_wait_asynccnt 1" ::: "memory");
}

__device__ __forceinline__ uint32_t lds_off_u32(const float* p) {
    // low 32 bits of the generic address of a __shared__ object are the LDS byte offset
    return (uint32_t)(uintptr_t)p;
}

__device__ __forceinline__ float clip30(float v) {
    return fminf(fmaxf(v, -SINH_CLIP_C), SINH_CLIP_C);
}

// ---------- block tree reduce (NT threads, NT a power of two) ----------
template<int NT>
__device__ __forceinline__ float block_sum(float v, float* red) {
    const int t = threadIdx.x;
    red[t] = v; __syncthreads();
#pragma unroll
    for (int o = NT >> 1; o > 0; o >>= 1) {
        if (t < o) red[t] += red[t + o];
        __syncthreads();
    }
    float r = red[0]; __syncthreads();
    return r;
}

// ================= init: mean0 = x[0], weight_pt = expmap0(weight) ===========
__global__ void k_init(const float* __restrict__ x, const float* __restrict__ weight,
                       const float* __restrict__ post_gain,
                       float* __restrict__ mean, float* __restrict__ wp,
                       float* __restrict__ scal) {
    const int lane = threadIdx.x;                 // 32 threads == 1 wave
    const float4 m4 = *(const float4*)(x + 4 * lane);
    *(float4*)(mean + 4 * lane) = m4;

    const float4 w4 = *(const float4*)(weight + 4 * lane);
    const float n2 = wred(w4.x * w4.x + w4.y * w4.y + w4.z * w4.z + w4.w * w4.w);
    const float n  = sqrtf(n2);
    const float c  = (n <= EPS_C) ? 0.0f : (sinhf(n) / fmaxf(n, EPS_C));
    float4 p4; p4.x = c * w4.x; p4.y = c * w4.y; p4.z = c * w4.z; p4.w = c * w4.w;
    *(float4*)(wp + 4 * lane) = p4;

    if (lane == 0) {
        scal[0] = 0.0f;                                            // done flag
        scal[2] = fminf(fmaxf(post_gain[0], 0.5f), 3.0f);          // gain
        scal[3] = c * c * n2;                                      // |wp|^2
    }
}

// ======= per-iteration partial sums of log_map(mean, x_i) over samples =======
__global__ __launch_bounds__(BLOCK)
void k_step_partials(const float* __restrict__ x, const float* __restrict__ meanp,
                     float* __restrict__ partials, int N, int spw) {
    __shared__ __align__(16) float sbuf[WPB * 2 * 128];  // double buffer per wave
    __shared__ __align__(16) float sP[WPB * 128];        // per-wave partial vectors

    const int lane = threadIdx.x & 31;
    const int w    = threadIdx.x >> 5;

    const float4 m4 = *(const float4*)(meanp + 4 * lane);
    const float mn2 = wred(m4.x * m4.x + m4.y * m4.y + m4.z * m4.z + m4.w * m4.w);
    const float bu  = 1.0f / sqrtf(1.0f + mn2);
    const float Ac  = bu / (1.0f + bu);

    const int  gw  = blockIdx.x * WPB + w;
    const long s0  = (long)gw * spw;
    int cnt = (int)((long)N - s0); if (cnt > spw) cnt = spw; if (cnt < 0) cnt = 0;

    const uint32_t ldsb = lds_off_u32(&sbuf[w * 256]) + 16u * (uint32_t)lane;

    float ax = 0.f, ay = 0.f, az = 0.f, aw = 0.f;
    if (cnt > 0) {
        async_row(ldsb, x + s0 * 128 + 4 * lane);
        for (int j = 0; j < cnt; ++j) {
            if (j + 1 < cnt) {
                async_row(ldsb + ((uint32_t)(j + 1) & 1u) * 512u,
                          x + (s0 + j + 1) * 128 + 4 * lane);
                wait_async_le(1);
            } else {
                wait_async_le(0);
            }
            const float4 xv = *(const float4*)(&sbuf[w * 256 + (j & 1) * 128 + 4 * lane]);
            const float dot_mx = wred(m4.x * xv.x + m4.y * xv.y + m4.z * xv.z + m4.w * xv.w);
            const float xn2    = wred(xv.x * xv.x + xv.y * xv.y + xv.z * xv.z + xv.w * xv.w);
            const float bv  = 1.0f / sqrtf(1.0f + xn2);
            const float c   = Ac * (-dot_mx) + (1.0f - bv) / bv;
            const float opc = 1.0f + c;                 // w = x - opc*mean
            const float wn2 = fmaxf(xn2 - 2.0f * opc * dot_mx + opc * opc * mn2, 0.0f);
            const float n   = sqrtf(wn2);
            const float cf  = (n <= EPS_C) ? 0.0f : (asinhf(n) / n);
            const float cm  = cf * opc;
            ax += cf * xv.x - cm * m4.x;
            ay += cf * xv.y - cm * m4.y;
            az += cf * xv.z - cm * m4.z;
            aw += cf * xv.w - cm * m4.w;
        }
    }
    sP[w * 128 + 4 * lane + 0] = ax;
    sP[w * 128 + 4 * lane + 1] = ay;
    sP[w * 128 + 4 * lane + 2] = az;
    sP[w * 128 + 4 * lane + 3] = aw;
    __syncthreads();

    // ---- block reduction of 8 wave-partials via V_WMMA_F32_16X16X4_F32 ----
    // A (16x4 f32, documented layout): rows m = 16 dims of a chunk, k = wave idx.
    // B = all-ones (value-identical under any lane layout) => D[m][*] = sum_k A[m][k].
    if (threadIdx.x < 32) {
        float* outp = partials + (long)blockIdx.x * 128;
        const int m  = lane & 15;
        const int kb = (lane >> 4) * 2;   // lanes 0-15 -> k 0,1 ; lanes 16-31 -> k 2,3
        v2f bones; bones.x = 1.0f; bones.y = 1.0f;
#pragma unroll
        for (int ch = 0; ch < 8; ++ch) {
            const int d0 = ch * 16 + m;
            v2f a0, a1;
            a0.x = sP[(kb + 0) * 128 + d0];  a0.y = sP[(kb + 1) * 128 + d0];  // waves 0..3
            a1.x = sP[(kb + 4) * 128 + d0];  a1.y = sP[(kb + 5) * 128 + d0];  // waves 4..7
            v8f cacc = {};
            cacc = __builtin_amdgcn_wmma_f32_16x16x4_f32(false, a0, false, bones,
                                                         (short)0, cacc, false, false);
            cacc = __builtin_amdgcn_wmma_f32_16x16x4_f32(false, a1, false, bones,
                                                         (short)0, cacc, false, false);
            // D layout: lanes 0-15 hold rows 0..7 in regs 0..7; lanes 16-31 rows 8..15
            if (lane == 0) {
#pragma unroll
                for (int r = 0; r < 8; ++r) outp[ch * 16 + r] = cacc[r];
            }
            if (lane == 16) {
#pragma unroll
                for (int r = 0; r < 8; ++r) outp[ch * 16 + 8 + r] = cacc[r];
            }
        }
    }
}

// ============ finalize one Frechet iteration (single block, 128 thr) =========
__global__ void k_step_finalize(float* __restrict__ meanp, const float* __restrict__ partials,
                                float* __restrict__ scal, int N) {
    __shared__ float red[128];
    const int t = threadIdx.x;                      // 128 threads, t == dim
    const float doneF = scal[0];                    // read BEFORE any flag update

    float s = 0.0f;
    for (int b = 0; b < NBLK; ++b) s += partials[b * 128 + t];
    float step = s / (float)N;

    const float sn2   = block_sum<128>(step * step, red);
    const float nrm   = sqrtf(sn2);
    const float sn    = fmaxf(nrm, 1e-8f);
    const float scale = fminf(MAX_STEP_C / sn, 1.0f);
    step *= scale;
    const float ns = nrm * scale;

    const float ec = (ns <= EPS_C) ? 0.0f : (sinhf(ns) / fmaxf(ns, EPS_C));
    const float e  = ec * step;                     // expmap0(step)
    const float en2 = ec * ec * ns * ns;

    const float mean   = meanp[t];
    const float mn2    = block_sum<128>(mean * mean, red);
    const float dot_me = block_sum<128>(mean * e, red);

    const float bu = 1.0f / sqrtf(1.0f + mn2);
    const float bv = 1.0f / sqrtf(1.0f + en2);
    const float c2 = bu / (1.0f + bu) * dot_me + (1.0f - bv) / bv;
    const float nm = mean + e + c2 * mean;          // gyro_add(mean, e)

    const float diff2 = block_sum<128>((nm - mean) * (nm - mean), red);

    const bool done = (doneF != 0.0f);
    if (!done) meanp[t] = nm;
    if (t == 0 && !done && diff2 < TOL_C * TOL_C) scal[0] = 1.0f;
}

// ================= variance partials: sum of dist(mean,x)^2 ==================
__global__ __launch_bounds__(BLOCK)
void k_var_partials(const float* __restrict__ x, const float* __restrict__ meanp,
                    float* __restrict__ vpart, int N, int spw) {
    __shared__ __align__(16) float sbuf[WPB * 2 * 128];
    __shared__ float sD[WPB];

    const int lane = threadIdx.x & 31;
    const int w    = threadIdx.x >> 5;

    const float4 m4 = *(const float4*)(meanp + 4 * lane);
    const float mn2 = wred(m4.x * m4.x + m4.y * m4.y + m4.z * m4.z + m4.w * m4.w);
    const float bu  = 1.0f / sqrtf(1.0f + mn2);
    const float Ac  = bu / (1.0f + bu);

    const int  gw = blockIdx.x * WPB + w;
    const long s0 = (long)gw * spw;
    int cnt = (int)((long)N - s0); if (cnt > spw) cnt = spw; if (cnt < 0) cnt = 0;

    const uint32_t ldsb = lds_off_u32(&sbuf[w * 256]) + 16u * (uint32_t)lane;

    float acc = 0.0f;
    if (cnt > 0) {
        async_row(ldsb, x + s0 * 128 + 4 * lane);
        for (int j = 0; j < cnt; ++j) {
            if (j + 1 < cnt) {
                async_row(ldsb + ((uint32_t)(j + 1) & 1u) * 512u,
                          x + (s0 + j + 1) * 128 + 4 * lane);
                wait_async_le(1);
            } else {
                wait_async_le(0);
            }
            const float4 xv = *(const float4*)(&sbuf[w * 256 + (j & 1) * 128 + 4 * lane]);
            const float dot_mx = wred(m4.x * xv.x + m4.y * xv.y + m4.z * xv.z + m4.w * xv.w);
            const float xn2    = wred(xv.x * xv.x + xv.y * xv.y + xv.z * xv.z + xv.w * xv.w);
            const float bv  = 1.0f / sqrtf(1.0f + xn2);
            const float c   = Ac * (-dot_mx) + (1.0f - bv) / bv;
            const float opc = 1.0f + c;
            const float wn2 = fmaxf(xn2 - 2.0f * opc * dot_mx + opc * opc * mn2, 0.0f);
            const float d   = asinhf(sqrtf(wn2));
            acc += d * d;
        }
    }
    if (lane == 0) sD[w] = acc;
    __syncthreads();
    if (threadIdx.x == 0) {
        float sum = 0.0f;
#pragma unroll
        for (int i = 0; i < WPB; ++i) sum += sD[i];
        vpart[blockIdx.x] = sum;
    }
}

// ============== finalize variance -> factor (single block, 256 thr) ==========
__global__ void k_var_finalize(const float* __restrict__ vpart, const float* __restrict__ shift,
                               float* __restrict__ scal, int N) {
    __shared__ float red[NBLK];
    const float tot = block_sum<NBLK>(vpart[threadIdx.x], red);
    if (threadIdx.x == 0) {
        float var = tot / (float)N;
        if (!(var == var)) var = 0.0f;               // nan_to_num
        var = fmaxf(var, 1e-8f);
        scal[1] = shift[0] / sqrtf(fmaxf(var, VAR_FLOOR_C) + BN_EPS_C);
    }
}

// ======================= output pass =========================================
__global__ __launch_bounds__(BLOCK)
void k_out(const float* __restrict__ x, float* __restrict__ out,
           const float* __restrict__ meanp, const float* __restrict__ wpp,
           const float* __restrict__ scal, int N, int spw) {
    __shared__ __align__(16) float sbuf[WPB * 2 * 128];

    const int lane = threadIdx.x & 31;
    const int w    = threadIdx.x >> 5;

    const float4 m4 = *(const float4*)(meanp + 4 * lane);
    const float4 p4 = *(const float4*)(wpp + 4 * lane);
    const float mn2  = wred(m4.x * m4.x + m4.y * m4.y + m4.z * m4.z + m4.w * m4.w);
    const float wpn2 = wred(p4.x * p4.x + p4.y * p4.y + p4.z * p4.z + p4.w * p4.w);
    const float bu  = 1.0f / sqrtf(1.0f + mn2);
    const float Ac  = bu / (1.0f + bu);
    const float buw = 1.0f / sqrtf(1.0f + wpn2);
    const float Bc  = buw / (1.0f + buw);
    const float factor = scal[1];
    const float gain   = scal[2];

    const int  gw = blockIdx.x * WPB + w;
    const long s0 = (long)gw * spw;
    int cnt = (int)((long)N - s0); if (cnt > spw) cnt = spw; if (cnt < 0) cnt = 0;

    const uint32_t ldsb = lds_off_u32(&sbuf[w * 256]) + 16u * (uint32_t)lane;

    if (cnt > 0) {
        async_row(ldsb, x + s0 * 128 + 4 * lane);
        for (int j = 0; j < cnt; ++j) {
            if (j + 1 < cnt) {
                async_row(ldsb + ((uint32_t)(j + 1) & 1u) * 512u,
                          x + (s0 + j + 1) * 128 + 4 * lane);
                wait_async_le(1);
            } else {
                wait_async_le(0);
            }
            const float4 xv = *(const float4*)(&sbuf[w * 256 + (j & 1) * 128 + 4 * lane]);
            const float dot_mx = wred(m4.x * xv.x + m4.y * xv.y + m4.z * xv.z + m4.w * xv.w);
            const float xn2    = wred(xv.x * xv.x + xv.y * xv.y + xv.z * xv.z + xv.w * xv.w);
            const float bv  = 1.0f / sqrtf(1.0f + xn2);
            const float c   = Ac * (-dot_mx) + (1.0f - bv) / bv;
            const float opc = 1.0f + c;
            float wx = xv.x - opc * m4.x, wy = xv.y - opc * m4.y;
            float wz = xv.z - opc * m4.z, ww = xv.w - opc * m4.w;
            const float wn2 = fmaxf(xn2 - 2.0f * opc * dot_mx + opc * opc * mn2, 0.0f);
            const float n   = sqrtf(wn2);
            // pv_gyro_scalar_mul(w, factor)
            const float ra = clip30(factor * asinhf(n));
            const float sc = (n <= EPS_C) ? 0.0f : (sinhf(ra) / fmaxf(n, EPS_C));
            const float yx = sc * wx, yy = sc * wy, yz = sc * wz, yw = sc * ww;
            const float yn2 = sc * sc * wn2;
            // gyro_add(weight_pt, y)
            const float dot_wy = wred(p4.x * yx + p4.y * yy + p4.z * yz + p4.w * yw);
            const float bvy = 1.0f / sqrtf(1.0f + yn2);
            const float c3  = Bc * dot_wy + (1.0f - bvy) / bvy;
            const float zx = p4.x + yx + c3 * p4.x;
            const float zy = p4.y + yy + c3 * p4.y;
            const float zz = p4.z + yz + c3 * p4.z;
            const float zw = p4.w + yw + c3 * p4.w;
            const float zn2 = wred(zx * zx + zy * zy + zz * zz + zw * zw);
            // pv_gyro_scalar_mul(z, gain)
            const float nz  = sqrtf(zn2);
            const float raz = clip30(gain * asinhf(nz));
            const float sc2 = (nz <= EPS_C) ? 0.0f : (sinhf(raz) / fmaxf(nz, EPS_C));
            float4 o4; o4.x = sc2 * zx; o4.y = sc2 * zy; o4.z = sc2 * zz; o4.w = sc2 * zw;
            *(float4*)(out + (s0 + j) * 128 + 4 * lane) = o4;
        }
    }
}

// ============================== host side ====================================
extern "C" void kernel_launch(void* const* d_in, const int* in_sizes, int n_in,
                              void* d_out, int out_size, void* d_ws, size_t ws_size,
                              hipStream_t stream) {
    const float* x         = (const float*)d_in[0];
    const float* weight    = (const float*)d_in[1];
    const float* shift     = (const float*)d_in[2];
    const float* post_gain = (const float*)d_in[3];
    float* out = (float*)d_out;
    float* ws  = (float*)d_ws;

    const int N = in_sizes[0] / 128;

    float* mean     = ws;                 // 128
    float* wp       = ws + 128;           // 128
    float* scal     = ws + 256;           // 32 scalars
    float* vpart    = ws + 288;           // NBLK
    float* partials = ws + 288 + NBLK;    // NBLK*128

    const int totalWaves = NBLK * WPB;
    const int spw = (N + totalWaves - 1) / totalWaves;

    k_init<<<1, 32, 0, stream>>>(x, weight, post_gain, mean, wp, scal);
    for (int it = 0; it < 8; ++it) {
        k_step_partials<<<NBLK, BLOCK, 0, stream>>>(x, mean, partials, N, spw);
        k_step_finalize<<<1, 128, 0, stream>>>(mean, partials, scal, N);
    }
    k_var_partials<<<NBLK, BLOCK, 0, stream>>>(x, mean, vpart, N, spw);
    k_var_finalize<<<1, NBLK, 0, stream>>>(vpart, shift, scal, N);
    k_out<<<NBLK, BLOCK, 0, stream>>>(x, out, mean, wp, scal, N, spw);
}